// Ctno_64269890618038
// MI455X (gfx1250) — compile-verified
//
#include <hip/hip_runtime.h>

// ---------------------------------------------------------------------------
// CTNO long-conv via FFT on MI455X (gfx1250, wave32).
//   out[b,h,t,d] = (x[b,h,:,d] (*) a[h,:,d])[t],  a = decay * sum_k coef*cos
// rfft8192 = packed complex fft4096, four-step 64x64 using
// V_WMMA_F32_16X16X4_F32 complex matmuls. Filter spectra V[h,f,d] precomputed
// into d_ws (H*D*4097 complex = 16.8MB, L2-resident in the 192MB L2).
//
// LDS layout uses stride-65 padding (plin(f) = f + f/64) so every ds access
// (fragment loads, twiddle, tile stores, pointwise) is bank-conflict-free on
// the 64-bank LDS; the DFT matrix is read transposed (it is symmetric) for
// the same reason. x is packed into LDS with global_load_async_to_lds_b32
// (ASYNCcnt) since its per-element stride (256B) defeats coalesced VMEM.
// ---------------------------------------------------------------------------

#define BB 4
#define HH 8
#define NN 4096
#define DD 64
#define KK 16
#define NC 4096           // packed complex FFT length (2N / 2)
#define THREADS 256       // 8 wave32s
#define PST 65            // padded row stride (64 banks + 1)

typedef __attribute__((ext_vector_type(2))) float v2f;
typedef __attribute__((ext_vector_type(8))) float v8f;

__device__ __forceinline__ int plin(int f) { return f + (f >> 6); }

struct cfx { float re, im; };
__device__ __forceinline__ cfx cmulx(cfx a, cfx b) {
  return cfx{a.re * b.re - a.im * b.im, a.re * b.im + a.im * b.re};
}

// ---------------------------------------------------------------------------
// 64x64x64 complex matmul C = A @ B with V_WMMA_F32_16X16X4_F32.
// AT=1 reads A transposed (valid for symmetric F64), strides parameterized so
// every LDS access lands on distinct banks. 8 waves x 2 tiles, K in 16 steps.
// A-frag (16x4 f32): lanes 0-15 K=k0*4+{0,1}, lanes 16-31 K=+{2,3}.
// C/D frag: VGPR j: lanes0-15 M=j, lanes16-31 M=j+8.
// ---------------------------------------------------------------------------
template <int AT, int ASTR, int BSTR, int DRS, int DCS>
__device__ __forceinline__ void cmatmul64(const float* Are, const float* Aim,
                                          const float* Bre, const float* Bim,
                                          float* Cre, float* Cim, int tid) {
  const int wid  = tid >> 5;
  const int lane = tid & 31;
  const int half = lane >> 4;
  const int l15  = lane & 15;

  for (int rep = 0; rep < 2; ++rep) {
    const int tile = wid + rep * 8;     // 0..15
    const int mt = tile >> 2;
    const int nt = tile & 3;
    const int mrow = mt * 16 + l15;
    const int ncol = nt * 16 + l15;

    v8f cre = {};
    v8f cim = {};
    for (int k0 = 0; k0 < 16; ++k0) {
      const int kb = k0 * 4 + half * 2;
      v2f ar, ai, nai, br, bi;
      if (AT) {
        ar[0] = Are[kb * ASTR + mrow];       ar[1] = Are[(kb + 1) * ASTR + mrow];
        ai[0] = Aim[kb * ASTR + mrow];       ai[1] = Aim[(kb + 1) * ASTR + mrow];
      } else {
        ar[0] = Are[mrow * ASTR + kb];       ar[1] = Are[mrow * ASTR + kb + 1];
        ai[0] = Aim[mrow * ASTR + kb];       ai[1] = Aim[mrow * ASTR + kb + 1];
      }
      br[0] = Bre[kb * BSTR + ncol];         br[1] = Bre[(kb + 1) * BSTR + ncol];
      bi[0] = Bim[kb * BSTR + ncol];         bi[1] = Bim[(kb + 1) * BSTR + ncol];
      nai[0] = -ai[0];                       nai[1] = -ai[1];
      // Cre += Ar*Br - Ai*Bi ; Cim += Ar*Bi + Ai*Br
      cre = __builtin_amdgcn_wmma_f32_16x16x4_f32(false, ar,  false, br,
                                                  (short)0, cre, false, false);
      cre = __builtin_amdgcn_wmma_f32_16x16x4_f32(false, nai, false, bi,
                                                  (short)0, cre, false, false);
      cim = __builtin_amdgcn_wmma_f32_16x16x4_f32(false, ar,  false, bi,
                                                  (short)0, cim, false, false);
      cim = __builtin_amdgcn_wmma_f32_16x16x4_f32(false, ai,  false, br,
                                                  (short)0, cim, false, false);
    }
#pragma unroll
    for (int j = 0; j < 8; ++j) {
      const int m = mt * 16 + j + half * 8;
      Cre[m * DRS + ncol * DCS] = cre[j];
      Cim[m * DRS + ncol * DCS] = cim[j];
    }
  }
}

// ---------------------------------------------------------------------------
// In-place 4096-pt complex FFT (four-step 64x64).  Input/output in Z with the
// padded-linear layout plin(). Output logical index f=k1+64*k2 is stored by
// stage 2 directly at plin(f) = m + 65*ncol (no separate transpose pass).
// ---------------------------------------------------------------------------
__device__ void fft4096(float* zre, float* zim, float* spre, float* spim,
                        const float* fre, const float* fim, int tid) {
  __syncthreads();
  // stage 1: SP(65-stride) = F @ Zmat ; A = F read transposed (symmetric)
  cmatmul64<1, 64, PST, PST, 1>(fre, fim, zre, zim, spre, spim, tid);
  __syncthreads();
  for (int i = tid; i < 4096; i += THREADS) {            // SP .*= W4096^{k1*n2}
    const int k1 = i >> 6, n2 = i & 63;
    const int idx = k1 * PST + n2;
    const float ang = -6.283185307179586f * (float)((k1 * n2) & 4095) / 4096.0f;
    float s, c; __sincosf(ang, &s, &c);
    const float xr = spre[idx], xi = spim[idx];
    spre[idx] = xr * c - xi * s;
    spim[idx] = xr * s + xi * c;
  }
  __syncthreads();
  // stage 2: Z[plin(k1+64*k2)] = (SP @ F)[k1][k2]
  cmatmul64<0, PST, 64, 1, PST>(spre, spim, fre, fim, zre, zim, tid);
  __syncthreads();
}

__device__ __forceinline__ void init_f64(float* fre, float* fim, int tid) {
  for (int i = tid; i < 4096; i += THREADS) {
    const int r = i >> 6, c = i & 63;
    const float ang = -6.283185307179586f * (float)((r * c) & 63) / 64.0f;
    float s, co; __sincosf(ang, &s, &co);
    fre[i] = co; fim[i] = s;
  }
}

// Packed-real unpack: E[f], E[(4096-f)&4095] -> rfft values X[f], X[4096-f].
__device__ __forceinline__ void unpack_pair(cfx Ef, cfx Eg, int f,
                                            cfx* Xf, cfx* Xp) {
  const float ang = -3.14159265358979f * (float)f / 4096.0f;   // Wf
  float s, c; __sincosf(ang, &s, &c);
  cfx A  = {0.5f * (Ef.re + Eg.re), 0.5f * (Ef.im - Eg.im)};
  cfx Bv = {0.5f * (Ef.re - Eg.re), 0.5f * (Ef.im + Eg.im)};
  cfx t = cmulx(cfx{s, -c}, Bv);                               // -i*Wf
  Xf->re = A.re + t.re; Xf->im = A.im + t.im;
  cfx A2 = {0.5f * (Eg.re + Ef.re), 0.5f * (Eg.im - Ef.im)};
  cfx B2 = {0.5f * (Eg.re - Ef.re), 0.5f * (Eg.im + Ef.im)};
  cfx t2 = cmulx(cfx{s, c}, B2);                               // -i*(-conj(Wf))
  Xp->re = A2.re + t2.re; Xp->im = A2.im + t2.im;
}

// Repack: Y[f], Y[4096-f] -> packed spectra E'[f], E'[4096-f] for the ifft.
__device__ __forceinline__ void repack_pair(cfx Yf, cfx Yp, int f,
                                            cfx* Ef, cfx* Eg) {
  const float ang = 3.14159265358979f * (float)f / 4096.0f;    // conj(Wf)
  float s, c; __sincosf(ang, &s, &c);
  cfx A  = {0.5f * (Yf.re + Yp.re), 0.5f * (Yf.im - Yp.im)};
  cfx Bv = {0.5f * (Yf.re - Yp.re), 0.5f * (Yf.im + Yp.im)};
  cfx O  = cmulx(cfx{c, s}, Bv);
  Ef->re = A.re - O.im; Ef->im = A.im + O.re;
  cfx A2 = {0.5f * (Yp.re + Yf.re), 0.5f * (Yp.im - Yf.im)};
  cfx B2 = {0.5f * (Yp.re - Yf.re), 0.5f * (Yp.im + Yf.im)};
  cfx O2 = cmulx(cfx{-c, s}, B2);
  Eg->re = A2.re - O2.im; Eg->im = A2.im + O2.re;
}

#define LDS_FLOATS (2 * 4096 + 4 * (64 * PST))

// ---------------------------------------------------------------------------
// Kernel 1: filter spectra  V[h,f,d] = rfft8192(decay * sum_k coef*cos)
// ---------------------------------------------------------------------------
__global__ void __launch_bounds__(THREADS)
ctno_filter_spectrum(const float* __restrict__ decay,
                     const float* __restrict__ cosb,
                     const float* __restrict__ coef,
                     float2* __restrict__ V) {
  __shared__ float lds[LDS_FLOATS];
  float* fre  = lds;          float* fim  = lds + 4096;
  float* zre  = lds + 8192;   float* zim  = zre + 64 * PST;
  float* spre = zim + 64 * PST;  float* spim = spre + 64 * PST;

  const int tid = threadIdx.x;
  const int d = blockIdx.x & 63;
  const int h = blockIdx.x >> 6;

  init_f64(fre, fim, tid);

  float cw[KK];
#pragma unroll
  for (int k = 0; k < KK; ++k) cw[k] = coef[(h * KK + k) * DD + d];

  for (int j = tid; j < 4096; j += THREADS) {   // pack z[j]=a[2j]+i*a[2j+1]
    const int t0 = 2 * j, t1 = 2 * j + 1;
    float s0 = 0.0f, s1 = 0.0f;
#pragma unroll
    for (int k = 0; k < KK; ++k) {
      s0 += cw[k] * cosb[t0 * KK + k];
      s1 += cw[k] * cosb[t1 * KK + k];
    }
    zre[plin(j)] = decay[t0] * s0;
    zim[plin(j)] = decay[t1] * s1;
  }

  fft4096(zre, zim, spre, spim, fre, fim, tid);

  float2* vb = V + (size_t)(h * DD + d) * (NC + 1);
  for (int f = tid; f <= 2048; f += THREADS) {
    const int g = (4096 - f) & 4095;
    cfx Ef = {zre[plin(f)], zim[plin(f)]};
    cfx Eg = {zre[plin(g)], zim[plin(g)]};
    cfx Xf, Xp;
    unpack_pair(Ef, Eg, f, &Xf, &Xp);
    vb[f] = make_float2(Xf.re, Xf.im);
    vb[4096 - f] = make_float2(Xp.re, Xp.im);
  }
}

// ---------------------------------------------------------------------------
// Kernel 2: per (b,h,d) channel: fwd FFT -> pointwise * V -> inv FFT -> store
// ---------------------------------------------------------------------------
__global__ void __launch_bounds__(THREADS)
ctno_fft_conv(const float* __restrict__ x,
              const float2* __restrict__ V,
              float* __restrict__ out) {
  __shared__ float lds[LDS_FLOATS];
  float* fre  = lds;          float* fim  = lds + 4096;
  float* zre  = lds + 8192;   float* zim  = zre + 64 * PST;
  float* spre = zim + 64 * PST;  float* spim = spre + 64 * PST;

  const int tid = threadIdx.x;
  const int d = blockIdx.x & 63;
  const int h = (blockIdx.x >> 6) & 7;
  const int b = blockIdx.x >> 9;

  init_f64(fre, fim, tid);

  // pack x column into LDS via async gather (stride-256B elements): the async
  // pipe (ASYNCcnt) writes LDS directly, no VGPR staging.
  const float* xcol = x + ((size_t)(b * HH + h) * NN) * DD + d;
  for (int j = tid; j < 2048; j += THREADS) {
    unsigned lre = (unsigned)(uintptr_t)(zre + plin(j));
    unsigned lim = (unsigned)(uintptr_t)(zim + plin(j));
    unsigned long long gre = (unsigned long long)(uintptr_t)(xcol + (size_t)(2 * j) * DD);
    unsigned long long gim = (unsigned long long)(uintptr_t)(xcol + (size_t)(2 * j + 1) * DD);
    asm volatile("global_load_async_to_lds_b32 %0, %1, off"
                 :: "v"(lre), "v"(gre) : "memory");
    asm volatile("global_load_async_to_lds_b32 %0, %1, off"
                 :: "v"(lim), "v"(gim) : "memory");
  }
  for (int j = 2048 + tid; j < 4096; j += THREADS) {
    zre[plin(j)] = 0.0f; zim[plin(j)] = 0.0f;
  }
  asm volatile("s_wait_asynccnt 0x0" ::: "memory");

  fft4096(zre, zim, spre, spim, fre, fim, tid);

  // unpack -> multiply by filter spectrum -> repack (conjugated for inverse)
  const float2* vb = V + (size_t)(h * DD + d) * (NC + 1);
  for (int f = tid; f <= 2048; f += THREADS) {
    const int g = (4096 - f) & 4095;
    cfx Ef = {zre[plin(f)], zim[plin(f)]};
    cfx Eg = {zre[plin(g)], zim[plin(g)]};
    cfx Xf, Xp;
    unpack_pair(Ef, Eg, f, &Xf, &Xp);
    const float2 vf = vb[f];
    const float2 vp = vb[4096 - f];
    cfx Yf = cmulx(Xf, cfx{vf.x, vf.y});
    cfx Yp = cmulx(Xp, cfx{vp.x, vp.y});
    cfx Epf, Epg;
    repack_pair(Yf, Yp, f, &Epf, &Epg);
    zre[plin(f)] = Epf.re;  zim[plin(f)] = -Epf.im;   // conj for inverse
    if (f != 0 && f != 2048) { zre[plin(g)] = Epg.re;  zim[plin(g)] = -Epg.im; }
  }

  fft4096(zre, zim, spre, spim, fre, fim, tid);       // ifft = conj(fft(conj))/N

  float* ocol = out + ((size_t)(b * HH + h) * NN) * DD + d;
  const float inv = 1.0f / 4096.0f;
  for (int j = tid; j < 2048; j += THREADS) {
    ocol[(size_t)(2 * j) * DD]     = zre[plin(j)] * inv;
    ocol[(size_t)(2 * j + 1) * DD] = -zim[plin(j)] * inv;
  }
}

extern "C" void kernel_launch(void* const* d_in, const int* in_sizes, int n_in,
                              void* d_out, int out_size, void* d_ws, size_t ws_size,
                              hipStream_t stream) {
  const float* x     = (const float*)d_in[0];
  const float* decay = (const float*)d_in[1];
  const float* cosb  = (const float*)d_in[2];
  const float* coef  = (const float*)d_in[3];
  // d_in[4] (index) is identity arange -> first-N slice, handled implicitly.
  float2* V = (float2*)d_ws;   // H*D*(NC+1) complex floats = 16.8 MB scratch

  ctno_filter_spectrum<<<HH * DD, THREADS, 0, stream>>>(decay, cosb, coef, V);
  ctno_fft_conv<<<BB * HH * DD, THREADS, 0, stream>>>(x, V, (float*)d_out);
}